// DiceLoss_86517821215299
// MI455X (gfx1250) — compile-verified
//
#include <hip/hip_runtime.h>

// ---------------------------------------------------------------------------
// Dice loss, MI455X (gfx1250, wave32).
// Problem: pred1/pred2 (2,14,48,256,256) f32, target (2,48,256,256) int.
// Bandwidth-bound streaming reduction -> per-thread accumulators ->
// wave32 shuffle reduce -> LDS -> 160-float global workspace atomics ->
// final single-wave kernel using V_WMMA_F32_16X16X4_F32 (ones-matrix reduce).
// ---------------------------------------------------------------------------

#define NUM_ORGAN 13
#define EPSF 1e-5f

static constexpr int       CHANS  = 14;
static constexpr long long VOX    = 48LL * 256 * 256;   // 3,145,728 spatial voxels
static constexpr int       WSROW  = 80;                 // 5 groups * 16 padded organ slots
// ws layout per batch b: [0..15]=num1, [16..31]=psq1, [32..47]=num2, [48..63]=psq2, [64..79]=cnt

typedef __attribute__((ext_vector_type(2))) float v2f;
typedef __attribute__((ext_vector_type(8))) float v8f;

__device__ __forceinline__ float wave_reduce_sum(float x) {
#pragma unroll
  for (int s = 16; s > 0; s >>= 1) x += __shfl_xor(x, s, 32);
  return x;
}

__global__ void zero_ws_kernel(float* __restrict__ ws) {
  const int i = threadIdx.x;
  if (i < 2 * WSROW) ws[i] = 0.0f;
}

__global__ __launch_bounds__(256) void dice_partial_kernel(
    const float* __restrict__ p1, const float* __restrict__ p2,
    const int* __restrict__ tgt, float* __restrict__ ws) {
  __shared__ float lds[WSROW];
  const int tid = threadIdx.x;
  if (tid < WSROW) lds[tid] = 0.0f;
  __syncthreads();

  const int b = blockIdx.y;
  const long long pbase = (long long)b * CHANS * VOX;
  const long long tbase = (long long)b * VOX;

  float num1[NUM_ORGAN], psq1[NUM_ORGAN], num2[NUM_ORGAN], psq2[NUM_ORGAN], cnt[NUM_ORGAN];
#pragma unroll
  for (int j = 0; j < NUM_ORGAN; ++j) {
    num1[j] = 0.f; psq1[j] = 0.f; num2[j] = 0.f; psq2[j] = 0.f; cnt[j] = 0.f;
  }

  const long long nquad  = VOX >> 2;                                  // 786,432
  const long long stride = (long long)gridDim.x * blockDim.x;

  for (long long q = (long long)blockIdx.x * blockDim.x + tid; q < nquad; q += stride) {
    const long long v = q << 2;
    // Prefetch next iteration's first streams (gfx1250 global_prefetch_b8).
    const long long vn = v + (stride << 2);
    if (vn < VOX) {
      __builtin_prefetch(p1 + pbase + VOX + vn, 0, 1);
      __builtin_prefetch(p2 + pbase + VOX + vn, 0, 1);
    }

    const int4 t = *reinterpret_cast<const int4*>(tgt + tbase + v);

#pragma unroll
    for (int j = 0; j < NUM_ORGAN; ++j) {
      const long long off = pbase + (long long)(j + 1) * VOX + v;
      const float4 a = *reinterpret_cast<const float4*>(p1 + off);
      const float4 c = *reinterpret_cast<const float4*>(p2 + off);

      psq1[j] += a.x * a.x + a.y * a.y + a.z * a.z + a.w * a.w;
      psq2[j] += c.x * c.x + c.y * c.y + c.z * c.z + c.w * c.w;

      const int id = j + 1;
      const float m0 = (t.x == id) ? 1.0f : 0.0f;
      const float m1 = (t.y == id) ? 1.0f : 0.0f;
      const float m2 = (t.z == id) ? 1.0f : 0.0f;
      const float m3 = (t.w == id) ? 1.0f : 0.0f;

      num1[j] += a.x * m0 + a.y * m1 + a.z * m2 + a.w * m3;
      num2[j] += c.x * m0 + c.y * m1 + c.z * m2 + c.w * m3;
      cnt[j]  += m0 + m1 + m2 + m3;
    }
  }

  // wave32 shuffle reduction, then one LDS atomic per wave per accumulator.
  const bool lane0 = (tid & 31) == 0;
#pragma unroll
  for (int j = 0; j < NUM_ORGAN; ++j) {
    float r;
    r = wave_reduce_sum(num1[j]); if (lane0) atomicAdd(&lds[0 * 16 + j], r);
    r = wave_reduce_sum(psq1[j]); if (lane0) atomicAdd(&lds[1 * 16 + j], r);
    r = wave_reduce_sum(num2[j]); if (lane0) atomicAdd(&lds[2 * 16 + j], r);
    r = wave_reduce_sum(psq2[j]); if (lane0) atomicAdd(&lds[3 * 16 + j], r);
    r = wave_reduce_sum(cnt[j]);  if (lane0) atomicAdd(&lds[4 * 16 + j], r);
  }
  __syncthreads();

  if (tid < WSROW) atomicAdd(&ws[b * WSROW + tid], lds[tid]);
}

// Final reduction: one wave32. The 10 needed sums (2 batches x {num1,psq1,
// num2,psq2,cnt}) become rows of A (16x4 f32); B = ones. Four chained
// V_WMMA_F32_16X16X4_F32 sweep the 16 padded organ slots; D[m,*] = row sums.
__global__ __launch_bounds__(32) void dice_final_kernel(
    const float* __restrict__ ws, float* __restrict__ out) {
  __shared__ float red[8 * 32];

  const int lane = threadIdx.x;
  const int m    = lane & 15;   // A-matrix row
  const int half = lane >> 4;   // 0 -> K{0,1}, 1 -> K{2,3}

  // Row m -> (batch, group) base in ws; rows >= 10 masked to zero.
  const int   bb    = (m < 5) ? 0 : 1;
  const int   g     = (m < 5) ? m : (m - 5);
  const int   base  = bb * WSROW + g * 16;
  const float valid = (m < 10) ? 1.0f : 0.0f;

  v2f bones; bones.x = 1.0f; bones.y = 1.0f;   // all 64 elements of B = 1
  v8f acc = {};

#pragma unroll
  for (int chunk = 0; chunk < 4; ++chunk) {
    const int k0 = chunk * 4 + half * 2;       // this lane supplies A[m,k0],A[m,k0+1]
    v2f a;
    a.x = ws[base + k0 + 0] * valid;
    a.y = ws[base + k0 + 1] * valid;
    acc = __builtin_amdgcn_wmma_f32_16x16x4_f32(
        /*neg_a=*/false, a, /*neg_b=*/false, bones,
        /*c_mod=*/(short)0, acc, /*reuse_a=*/false, /*reuse_b=*/false);
  }

  // D layout: VGPR r, lanes 0-15 -> M=r, N=lane; lanes 16-31 -> M=r+8, N=lane-16.
#pragma unroll
  for (int r = 0; r < 8; ++r) red[r * 32 + lane] = acc[r];
  __syncthreads();

  if (lane == 0) {
    float s[10];
#pragma unroll
    for (int i = 0; i < 10; ++i)
      s[i] = (i < 8) ? red[i * 32 + 0] : red[(i - 8) * 32 + 16];

    const float eps13 = (float)NUM_ORGAN * EPSF;
    // s[b*5+0]=Σnum1, +1=Σpsq1, +2=Σnum2, +3=Σpsq2, +4=Σcnt
    const float d1b0 = (2.0f * s[0]) / (s[1] + s[4] + eps13);
    const float d2b0 = (2.0f * s[2]) / (s[3] + s[4] + eps13);
    const float d1b1 = (2.0f * s[5]) / (s[6] + s[9] + eps13);
    const float d2b1 = (2.0f * s[7]) / (s[8] + s[9] + eps13);
    out[0] = 0.5f * ((2.0f - d1b0 - d2b0) + (2.0f - d1b1 - d2b1));
  }
}

extern "C" void kernel_launch(void* const* d_in, const int* in_sizes, int n_in,
                              void* d_out, int out_size, void* d_ws, size_t ws_size,
                              hipStream_t stream) {
  const float* p1  = (const float*)d_in[0];
  const float* p2  = (const float*)d_in[1];
  const int*   tgt = (const int*)d_in[2];
  float*       out = (float*)d_out;
  float*       ws  = (float*)d_ws;

  zero_ws_kernel<<<1, 256, 0, stream>>>(ws);

  dim3 grid(768, 2, 1);   // 768 blocks/batch * 256 thr -> 4 float4-quads per thread
  dice_partial_kernel<<<grid, 256, 0, stream>>>(p1, p2, tgt, ws);

  dice_final_kernel<<<1, 32, 0, stream>>>(ws, out);
}